// Quantize2_43645457662411
// MI455X (gfx1250) — compile-verified
//
#include <hip/hip_runtime.h>

typedef __attribute__((ext_vector_type(2))) float v2f;
typedef __attribute__((ext_vector_type(8))) float v8f;

#define DIM    256
#define NEMB   1024
#define NTOK   65536          // 16*64*64
#define MTILE  128            // tokens per block (8 WMMA M-subtiles)
#define NCHUNK 128            // codes processed per accumulator pass
#define NBLK   (NTOK / MTILE) // 512
#define LDSTR  260            // padded LDS row stride (floats), bank-conflict-free
#define TILE_BYTES (MTILE * LDSTR * 4)  // 133120

#define Q_SIZE   (NTOK * DIM)        // 16777216
#define Q_OFF    0
#define DIFF_OFF (Q_SIZE)            // 16777216
#define IND_OFF  (Q_SIZE + 1)        // 16777217
#define DIST_OFF (Q_SIZE + 1 + NTOK) // 16842753

// ---------------------------------------------------------------------------
// Kernel 1: per-code squared norms  enorm[j] = sum_k embed[k][j]^2
// ---------------------------------------------------------------------------
__global__ void k_enorm(const float* __restrict__ embed, float* __restrict__ enorm) {
    int j = blockIdx.x * blockDim.x + threadIdx.x; // coalesced across codes
    float s = 0.f;
    for (int k = 0; k < DIM; ++k) {
        float e = embed[(size_t)k * NEMB + j];
        s = fmaf(e, e, s);
    }
    enorm[j] = s;
}

// ---------------------------------------------------------------------------
// Kernel 2: embedT[n][k] = embed[k][n]  (LDS-tiled transpose, 32x32 tiles)
// ---------------------------------------------------------------------------
__global__ void k_transpose(const float* __restrict__ embed, float* __restrict__ embedT) {
    __shared__ float tile[32][33];
    const int nT = blockIdx.x * 32; // column base in embed
    const int kT = blockIdx.y * 32; // row base in embed
    const int x = threadIdx.x, y = threadIdx.y; // blockDim (32,8)
    for (int yy = y; yy < 32; yy += 8)
        tile[yy][x] = embed[(size_t)(kT + yy) * NEMB + nT + x];
    __syncthreads();
    for (int yy = y; yy < 32; yy += 8)
        embedT[(size_t)(nT + yy) * DIM + kT + x] = tile[x][yy];
}

// ---------------------------------------------------------------------------
// Kernel 3: main VQ kernel. One block = 128 tokens x full 1024 codes,
//   processed as 8 N-chunks of 128 columns (accumulators recycled per chunk,
//   so embed is read from L2 exactly ONCE per block -> 512 MB total).
//   Wave w owns columns [chunk + w*16, +16) x all 128 rows:
//     - B fragment: 2 global loads per K-step, register-reused by 8 WMMAs
//     - A fragments: preloaded as an 8-entry array per K-step so the DS loads
//       form a clause and WMMAs retire with staggered s_wait_dscnt (ILP),
//       instead of one load->wait0->wmma chain per tile.
//   Running argmin: value in rv[8][8]; index compressed to a 4-bit chunk id
//   (n = c*128 + w*16 + lm is lane-derivable) -> stays under 256 VGPRs.
// ---------------------------------------------------------------------------
__global__ __launch_bounds__(256)
void k_vq(const float* __restrict__ x, const float* __restrict__ embed,
          const float* __restrict__ enorm, const float* __restrict__ embedT,
          float* __restrict__ qout, float* __restrict__ indout,
          float* __restrict__ distout, float* __restrict__ blocksum) {
    extern __shared__ float tile[];        // 128 x 256 input tile (padded rows)
    __shared__ float xnorm_s[MTILE];
    __shared__ float wminv[8][MTILE];
    __shared__ int   wmini[8][MTILE];
    __shared__ int   idx_s[MTILE];
    __shared__ float red[256];

    const int t  = threadIdx.x;
    const int tb = blockIdx.x * MTILE;

    // ---- stage input tile into LDS (float4, fully coalesced) ----
    #pragma unroll 4
    for (int i = 0; i < 32; ++i) {
        int j   = i * 256 + t;        // 8192 float4 total
        int row = j >> 6;             // 64 float4 per row
        int c4  = j & 63;
        const float4 v = ((const float4*)(x + (size_t)(tb + row) * DIM))[c4];
        *(float4*)&tile[row * LDSTR + c4 * 4] = v;
    }
    __syncthreads();

    // ---- per-token ||x||^2 ----
    if (t < MTILE) {
        float s = 0.f;
        for (int d = 0; d < DIM; ++d) {
            float v = tile[t * LDSTR + d];
            s = fmaf(v, v, s);
        }
        xnorm_s[t] = s;
    }
    __syncthreads();

    const int w    = t >> 5;
    const int lane = t & 31;
    const int lm   = lane & 15;       // M (A) / N (B,C) index within 16x16 tile
    const int hi   = lane >> 4;       // lane-half: selects K pair / M+8
    const int kl   = hi * 2;          // K offset within 4-wide slab

    // running per-lane argmin: value per (s,r); index as packed 4-bit chunk id
    float    rv[8][8];
    unsigned cid[8];
    #pragma unroll
    for (int s = 0; s < 8; ++s) {
        cid[s] = 0u;
        #pragma unroll
        for (int r = 0; r < 8; ++r) rv[s][r] = 3.0e38f;
    }

    v8f zero = {};
    const int abase = lm * LDSTR + kl;   // lane's LDS base within a row group

    for (int c = 0; c < NEMB / NCHUNK; ++c) {
        const int nb = c * NCHUNK + w * 16 + lm;   // this lane's column
        const float* bp = embed + (size_t)kl * NEMB + nb;

        v8f acc[8];
        #pragma unroll
        for (int s = 0; s < 8; ++s) acc[s] = zero;

        // ---- GEMM over K for this column group ----
        #pragma unroll 2
        for (int kb = 0; kb < DIM; kb += 4) {
            v2f b;  // B 4x16: VGPR = K within half, lane half = +2
            const size_t ko = (size_t)kb * NEMB;
            b.x = bp[ko];
            b.y = bp[ko + NEMB];

            // preload all 8 A fragments -> DS clause, overlapped with WMMAs
            v2f a[8];
            #pragma unroll
            for (int s = 0; s < 8; ++s) {
                a[s].x = tile[abase + s * (16 * LDSTR) + kb];
                a[s].y = tile[abase + s * (16 * LDSTR) + kb + 1];
            }
            #pragma unroll
            for (int s = 0; s < 8; ++s) {
                acc[s] = __builtin_amdgcn_wmma_f32_16x16x4_f32(
                    false, a[s], false, b, (short)0, acc[s], false, false);
            }
        }

        // ---- chunk epilogue: dist out + running min update ----
        // chunks ascend in n, so strict '<' keeps the first (lowest) index.
        const float en = enorm[nb];
        #pragma unroll
        for (int s = 0; s < 8; ++s) {
            #pragma unroll
            for (int r = 0; r < 8; ++r) {
                const int   mloc = s * 16 + hi * 8 + r;   // token row in block
                const float dval = xnorm_s[mloc] - 2.f * acc[s][r] + en;
                distout[(size_t)(tb + mloc) * NEMB + nb] = dval;
                if (dval < rv[s][r]) {
                    rv[s][r] = dval;
                    cid[s]   = (cid[s] & ~(0xFu << (4 * r))) |
                               ((unsigned)c << (4 * r));
                }
            }
        }
    }

    // ---- butterfly argmin within each 16-lane half (same token row) ----
    #pragma unroll
    for (int s = 0; s < 8; ++s) {
        #pragma unroll
        for (int r = 0; r < 8; ++r) {
            float v  = rv[s][r];
            int   id = (int)((cid[s] >> (4 * r)) & 0xFu) * NCHUNK + w * 16 + lm;
            #pragma unroll
            for (int off = 1; off <= 8; off <<= 1) {
                float ov = __shfl_xor(v, off, 32);
                int   oi = __shfl_xor(id, off, 32);
                if (ov < v || (ov == v && oi < id)) { v = ov; id = oi; }
            }
            if (lm == 0) { // lanes 0/16 publish per-token result for half
                wminv[w][s * 16 + hi * 8 + r] = v;
                wmini[w][s * 16 + hi * 8 + r] = id;
            }
        }
    }
    __syncthreads();

    // ---- cross-wave argmin (8 partials per token), first-index tie-break ----
    if (t < MTILE) {
        float bv = wminv[0][t];
        int   bi = wmini[0][t];
        for (int ww = 1; ww < 8; ++ww) {
            float cv = wminv[ww][t];
            int   ci = wmini[ww][t];
            if (cv < bv || (cv == bv && ci < bi)) { bv = cv; bi = ci; }
        }
        idx_s[t] = bi;
        indout[tb + t] = (float)bi;   // argmin index as f32 (exact, < 1024)
    }
    __syncthreads();

    // ---- gather quantize rows (coalesced via embedT) + diff partial ----
    float lsum = 0.f;
    #pragma unroll 4
    for (int i = 0; i < MTILE; ++i) {
        const int   ci = idx_s[i];
        const float q  = embedT[(size_t)ci * DIM + t];
        const float xv = tile[i * LDSTR + t];
        const float p  = q - xv;
        lsum = fmaf(p, p, lsum);
        qout[(size_t)(tb + i) * DIM + t] = q;
    }
    red[t] = lsum;
    __syncthreads();
    for (int s = 128; s > 0; s >>= 1) {
        if (t < s) red[t] += red[t + s];
        __syncthreads();
    }
    if (t == 0) blocksum[blockIdx.x] = red[0];
}

// ---------------------------------------------------------------------------
// Kernel 4: deterministic reduction of block partials -> diff scalar
// ---------------------------------------------------------------------------
__global__ void k_diff(const float* __restrict__ blocksum, float* __restrict__ diffout) {
    __shared__ float red[256];
    const int t = threadIdx.x;
    float s = 0.f;
    for (int i = t; i < NBLK; i += 256) s += blocksum[i];
    red[t] = s;
    __syncthreads();
    for (int k = 128; k > 0; k >>= 1) {
        if (t < k) red[t] += red[t + k];
        __syncthreads();
    }
    if (t == 0) diffout[0] = red[0] * (1.0f / (float)Q_SIZE);
}

// ---------------------------------------------------------------------------
extern "C" void kernel_launch(void* const* d_in, const int* in_sizes, int n_in,
                              void* d_out, int out_size, void* d_ws, size_t ws_size,
                              hipStream_t stream) {
    const float* x     = (const float*)d_in[0]; // [65536, 256]
    const float* embed = (const float*)d_in[1]; // [256, 1024]
    float* out = (float*)d_out;

    float* enorm    = (float*)d_ws;          // 1024 floats
    float* embedT   = enorm + NEMB;          // 1024*256 floats
    float* blocksum = embedT + NEMB * DIM;   // 512 floats

    k_enorm<<<NEMB / 256, 256, 0, stream>>>(embed, enorm);

    dim3 tpb(32, 8);
    dim3 grid(NEMB / 32, DIM / 32);
    k_transpose<<<grid, tpb, 0, stream>>>(embed, embedT);

    // allow >64KB dynamic LDS for the 133KB input tile (WGP has 320KB)
    (void)hipFuncSetAttribute((const void*)k_vq,
                              hipFuncAttributeMaxDynamicSharedMemorySize,
                              TILE_BYTES);

    k_vq<<<NBLK, 256, TILE_BYTES, stream>>>(x, embed, enorm, embedT,
                                            out + Q_OFF, out + IND_OFF,
                                            out + DIST_OFF, blocksum);

    k_diff<<<1, 256, 0, stream>>>(blocksum, out + DIFF_OFF);
}